// GAT_24309514895502
// MI455X (gfx1250) — compile-verified
//
#include <hip/hip_runtime.h>
#include <math.h>

// ---------------- problem constants (match reference) ----------------
#define NODES 50000
#define EDGES 800000
#define EP    (EDGES + NODES)   // edges + self loops = 850000
#define IN_F  256
#define HID   64
#define HEADS 8
#define OUT_F 40
#define LRELU 0.2f

typedef __attribute__((ext_vector_type(2))) float v2f;
typedef __attribute__((ext_vector_type(8))) float v8f;

// ---------------------------------------------------------------------
// Per-head fp32 WMMA GEMM: C[m, head*CHS + n] = A[m,:K] . B[head][:,n]
// Compile-time K/LDA/LDB/NCOLS -> immediate-offset clause loads, no runtime
// address mul in the loop, branchless out-of-range N handling (clamped B
// column; invalid lanes' WMMA output columns are simply never stored).
// V_WMMA_F32_16X16X4_F32 layouts (wave32):
//   A 16x4: lane<16 -> row=lane,    holds K=0,1 ; lane>=16 -> row=lane-16, K=2,3
//   B 4x16: lane<16 -> col=lane,    holds K=0,1 ; lane>=16 -> col=lane-16, K=2,3
//   C/D:    vgpr v: lanes 0-15 -> M=v, lanes 16-31 -> M=v+8 ; N=lane&15
// ---------------------------------------------------------------------
template <int K, int LDA, int LDB, int NCOLS, int CHS>
__global__ __launch_bounds__(128)
void gat_wmma_gemm(const float* __restrict__ A, const float* __restrict__ B,
                   float* __restrict__ C, int ldc)
{
    const int lane   = threadIdx.x & 31;
    const int wave   = threadIdx.x >> 5;
    const int head   = blockIdx.z;
    const int tile_m = blockIdx.x;
    const int tile_n = wave;                       // 4 waves cover 64 cols

    const int m  = tile_m * 16 + (lane & 15);
    const int n  = tile_n * 16 + (lane & 15);
    const int kb = (lane >> 4) * 2;                // 0 or 2

    // branchless in-bounds B column; out-of-range lanes compute garbage
    // in D columns that are never stored.
    constexpr bool full = (NCOLS % 16 == 0);
    const int  bn     = full ? n : (n < NCOLS ? n : 0);
    const bool nvalid = full ? true : (n < NCOLS);

    const float* __restrict__ Ar = A + (long)m * LDA;
    const float* __restrict__ Bc = B + (long)head * (K * LDB) + kb * LDB + bn;

    // warm WGP$/L2 for the streamed A row (global_prefetch_b8)
#pragma unroll
    for (int p = 0; p < K; p += 64)
        __builtin_prefetch(Ar + p, 0, 0);

    v8f acc = {};
#pragma unroll 4
    for (int k = 0; k < K; k += 4) {
        v2f a, b;
        a.x = Ar[k + kb];
        a.y = Ar[k + kb + 1];
        b.x = Bc[(long)k * LDB];
        b.y = Bc[(long)k * LDB + LDB];
        acc = __builtin_amdgcn_wmma_f32_16x16x4_f32(
                  false, a, false, b, (short)0, acc, false, false);
    }

    if (nvalid) {
        const int row0 = tile_m * 16 + ((lane >> 4) << 3);
        const long cofs = (long)head * CHS + n;
#pragma unroll
        for (int v = 0; v < 8; ++v)
            C[(long)(row0 + v) * ldc + cofs] = acc[v];
    }
}

// s_src[n,h] = Wh[n,h,:] . a[h][:F] ; s_dst[n,h] = Wh[n,h,:] . a[h][F:2F]
// float4 loads (F = 64 / 40, both multiples of 4; rows 16B-aligned).
__global__ void gat_scores(const float* __restrict__ Wh, const float* __restrict__ av,
                           float* __restrict__ s_src, float* __restrict__ s_dst,
                           int nheads, int F, int ldw, int n_nodes)
{
    int idx = blockIdx.x * blockDim.x + threadIdx.x;
    if (idx >= n_nodes * nheads) return;
    int node = idx / nheads;
    int h    = idx - node * nheads;
    const float4* w  = (const float4*)(Wh + (long)node * ldw + h * F);
    const float4* a0 = (const float4*)(av + (long)h * 2 * F);
    const float4* a1 = (const float4*)(av + (long)h * 2 * F + F);
    float ss = 0.0f, sd = 0.0f;
    for (int f = 0; f < F / 4; ++f) {
        float4 v = w[f], p = a0[f], q = a1[f];
        ss += v.x * p.x + v.y * p.y + v.z * p.z + v.w * p.w;
        sd += v.x * q.x + v.y * q.y + v.z * q.z + v.w * q.w;
    }
    s_src[idx] = ss;
    s_dst[idx] = sd;
}

__global__ void fill_f32(float* __restrict__ p, float v, long n)
{
    long i = (long)blockIdx.x * blockDim.x + threadIdx.x;
    if (i < n) p[i] = v;
}

// float atomic max via monotone integer atomics (no NaNs in this pipeline)
__device__ __forceinline__ void atomicMaxF32(float* addr, float val)
{
    if (val >= 0.0f) atomicMax((int*)addr, __float_as_int(val));
    else             atomicMin((unsigned int*)addr, __float_as_uint(val));
}

__device__ __forceinline__ void edge_rc(int e, const int* row, const int* col,
                                        int& r, int& c)
{
    if (e < EDGES) { r = row[e]; c = col[e]; }
    else           { r = e - EDGES; c = r; }           // appended self loops
}

__global__ void gat_edge_max(const int* __restrict__ row, const int* __restrict__ col,
                             const float* __restrict__ s_src, const float* __restrict__ s_dst,
                             float* __restrict__ mx, int nheads)
{
    long idx = (long)blockIdx.x * blockDim.x + threadIdx.x;
    if (idx >= (long)EP * nheads) return;
    int e = (int)(idx / nheads);
    int h = (int)(idx - (long)e * nheads);
    int r, c; edge_rc(e, row, col, r, c);
    float v = s_src[r * nheads + h] + s_dst[c * nheads + h];
    v = (v > 0.0f) ? v : LRELU * v;
    atomicMaxF32(&mx[c * nheads + h], v);
}

__global__ void gat_edge_expsum(const int* __restrict__ row, const int* __restrict__ col,
                                const float* __restrict__ s_src, const float* __restrict__ s_dst,
                                const float* __restrict__ mx,
                                float* __restrict__ ex, float* __restrict__ denom, int nheads)
{
    long idx = (long)blockIdx.x * blockDim.x + threadIdx.x;
    if (idx >= (long)EP * nheads) return;
    int e = (int)(idx / nheads);
    int h = (int)(idx - (long)e * nheads);
    int r, c; edge_rc(e, row, col, r, c);
    float v = s_src[r * nheads + h] + s_dst[c * nheads + h];
    v = (v > 0.0f) ? v : LRELU * v;
    float ev = __expf(v - mx[c * nheads + h]);
    ex[idx] = ev;
    atomicAdd(&denom[c * nheads + h], ev);
}

// one thread per (edge, head, feature); consecutive threads -> consecutive f
__global__ void gat_aggregate(const int* __restrict__ row, const int* __restrict__ col,
                              const float* __restrict__ Wh,
                              const float* __restrict__ ex, const float* __restrict__ denom,
                              float* __restrict__ out,
                              int nheads, int F, int ldw, int ldo)
{
    long idx = (long)blockIdx.x * blockDim.x + threadIdx.x;
    if (idx >= (long)EP * nheads * F) return;
    int  f  = (int)(idx % F);
    long eh = idx / F;
    int  e  = (int)(eh / nheads);
    int  h  = (int)(eh - (long)e * nheads);
    int r, c; edge_rc(e, row, col, r, c);
    float alpha = ex[eh] / denom[c * nheads + h];
    float val = Wh[(long)r * ldw + h * F + f] * alpha;
    atomicAdd(&out[(long)c * ldo + h * F + f], val);
}

__global__ void gat_elu(float* __restrict__ p, long n)
{
    long i = (long)blockIdx.x * blockDim.x + threadIdx.x;
    if (i < n) {
        float v = p[i];
        p[i] = (v > 0.0f) ? v : (__expf(v) - 1.0f);
    }
}

// final: ELU then log_softmax over OUT_F classes per node
__global__ void gat_final(const float* __restrict__ agg, float* __restrict__ out)
{
    int node = blockIdx.x * blockDim.x + threadIdx.x;
    if (node >= NODES) return;
    float v[OUT_F];
    float mx = -1e30f;
#pragma unroll
    for (int j = 0; j < OUT_F; ++j) {
        float t = agg[(long)node * OUT_F + j];
        t = (t > 0.0f) ? t : (__expf(t) - 1.0f);
        v[j] = t;
        mx = fmaxf(mx, t);
    }
    float s = 0.0f;
#pragma unroll
    for (int j = 0; j < OUT_F; ++j) s += __expf(v[j] - mx);
    float lse = mx + __logf(s);
#pragma unroll
    for (int j = 0; j < OUT_F; ++j)
        out[(long)node * OUT_F + j] = v[j] - lse;
}

// ---------------------------------------------------------------------
static inline unsigned int nblk(long n, int b) { return (unsigned int)((n + b - 1) / b); }

template <int K>
static void run_hidden_layer(const float* A,
                             const float* W, const float* avec,
                             const int* row, const int* col,
                             float* Wh, float* ssrc, float* sdst,
                             float* mx, float* dn, float* ex,
                             float* outagg, hipStream_t stream)
{
    dim3 gg(NODES / 16, 1, HEADS);
    gat_wmma_gemm<K, K, HID, HID, HID><<<gg, 128, 0, stream>>>(A, W, Wh, HEADS * HID);

    gat_scores<<<nblk((long)NODES * HEADS, 256), 256, 0, stream>>>(
        Wh, avec, ssrc, sdst, HEADS, HID, HEADS * HID, NODES);

    fill_f32<<<nblk((long)NODES * HEADS, 256), 256, 0, stream>>>(mx, -1e30f, (long)NODES * HEADS);
    fill_f32<<<nblk((long)NODES * HEADS, 256), 256, 0, stream>>>(dn, 0.0f, (long)NODES * HEADS);
    fill_f32<<<nblk((long)NODES * HEADS * HID, 256), 256, 0, stream>>>(
        outagg, 0.0f, (long)NODES * HEADS * HID);

    long em = (long)EP * HEADS;
    gat_edge_max<<<nblk(em, 256), 256, 0, stream>>>(row, col, ssrc, sdst, mx, HEADS);
    gat_edge_expsum<<<nblk(em, 256), 256, 0, stream>>>(row, col, ssrc, sdst, mx, ex, dn, HEADS);

    long ag = (long)EP * HEADS * HID;
    gat_aggregate<<<nblk(ag, 256), 256, 0, stream>>>(row, col, Wh, ex, dn, outagg,
                                                     HEADS, HID, HEADS * HID, HEADS * HID);
    gat_elu<<<nblk((long)NODES * HEADS * HID, 256), 256, 0, stream>>>(
        outagg, (long)NODES * HEADS * HID);
}

extern "C" void kernel_launch(void* const* d_in, const int* in_sizes, int n_in,
                              void* d_out, int out_size, void* d_ws, size_t ws_size,
                              hipStream_t stream)
{
    (void)in_sizes; (void)n_in; (void)out_size; (void)ws_size;

    const float* x    = (const float*)d_in[0];
    const int*   ei   = (const int*)  d_in[1];
    const float* W0   = (const float*)d_in[2];
    const float* a0   = (const float*)d_in[3];
    const float* W1   = (const float*)d_in[4];
    const float* a1   = (const float*)d_in[5];
    const float* Wout = (const float*)d_in[6];
    const float* aout = (const float*)d_in[7];
    float* out = (float*)d_out;

    const int* row = ei;            // edge_index[0]
    const int* col = ei + EDGES;    // edge_index[1]

    // -------- workspace carve-up (f32) --------
    float* ws = (float*)d_ws;
    size_t off = 0;
    float* Wh   = ws + off; off += (size_t)NODES * HEADS * HID;   // per-layer Wh
    float* H1   = ws + off; off += (size_t)NODES * HEADS * HID;   // layer-0 output
    float* H2   = ws + off; off += (size_t)NODES * HEADS * HID;   // layer-1 output
    float* ssrc = ws + off; off += (size_t)NODES * HEADS;
    float* sdst = ws + off; off += (size_t)NODES * HEADS;
    float* mx   = ws + off; off += (size_t)NODES * HEADS;
    float* dn   = ws + off; off += (size_t)NODES * HEADS;
    float* ex   = ws + off; off += (size_t)EP * HEADS;
    float* Who  = ws + off; off += (size_t)NODES * OUT_F;
    float* aggo = ws + off; off += (size_t)NODES * OUT_F;

    // -------- layer 0: x[N,256] -> H1[N,512] --------
    run_hidden_layer<IN_F>(x, W0, a0, row, col,
                           Wh, ssrc, sdst, mx, dn, ex, H1, stream);

    // -------- layer 1: H1[N,512] -> H2[N,512] --------
    run_hidden_layer<HEADS * HID>(H1, W1, a1, row, col,
                                  Wh, ssrc, sdst, mx, dn, ex, H2, stream);

    // -------- output layer: H2[N,512] -> aggo[N,40] --------
    dim3 go(NODES / 16, 1, 1);   // 4 waves cover n-tiles 0..3 (cols clamped to 40)
    gat_wmma_gemm<HEADS * HID, HEADS * HID, OUT_F, OUT_F, 0>
        <<<go, 128, 0, stream>>>(H2, Wout, Who, OUT_F);
    gat_scores<<<nblk(NODES, 256), 256, 0, stream>>>(Who, aout, ssrc, sdst, 1, OUT_F,
                                                     OUT_F, NODES);
    fill_f32<<<nblk(NODES, 256), 256, 0, stream>>>(mx, -1e30f, (long)NODES);
    fill_f32<<<nblk(NODES, 256), 256, 0, stream>>>(dn, 0.0f, (long)NODES);
    fill_f32<<<nblk((long)NODES * OUT_F, 256), 256, 0, stream>>>(aggo, 0.0f,
                                                                 (long)NODES * OUT_F);
    gat_edge_max<<<nblk(EP, 256), 256, 0, stream>>>(row, col, ssrc, sdst, mx, 1);
    gat_edge_expsum<<<nblk(EP, 256), 256, 0, stream>>>(row, col, ssrc, sdst, mx, ex, dn, 1);
    gat_aggregate<<<nblk((long)EP * OUT_F, 256), 256, 0, stream>>>(
        row, col, Who, ex, dn, aggo, 1, OUT_F, OUT_F, OUT_F);

    gat_final<<<nblk(NODES, 256), 256, 0, stream>>>(aggo, out);
}